// MinibatchDiscrimination_26895085207878
// MI455X (gfx1250) — compile-verified
//
#include <hip/hip_runtime.h>
#include <hip/hip_bf16.h>

// Problem constants from the reference
#define BATCH 1024
#define DDIM  1024
#define UNITS 16
#define ROW   8
#define NCOL  (UNITS * ROW)   // 128
#define OUTW  (DDIM + UNITS)  // 1040

typedef __attribute__((ext_vector_type(2))) float v2f;
typedef __attribute__((ext_vector_type(8))) float v8f;

// ---------------------------------------------------------------------------
// Kernel 1: Mg[b][u*8+r] = sum_d x[b][d] * T[u][d][r]
// Full-precision f32 GEMM using V_WMMA_F32_16X16X4_F32 (16x16 tile per wave).
//   A (16x4 f32): lane = {M = lane&15, K pair = 2*(lane>>4)}; VGPR0=K, VGPR1=K+1
//   B (4x16 f32): lane = {N = lane&15}; VGPR0=B[K][N], VGPR1=B[K+1][N]
//   C/D (16x16 f32, 8 VGPRs): VGPR v -> M = v + 8*(lane>>4), N = lane&15
// Grid: 64 blocks (row tiles), 256 threads = 8 waves (one 16-col tile each).
// ---------------------------------------------------------------------------
__global__ void __launch_bounds__(256)
mbd_gemm_wmma_f32(const float* __restrict__ x,   // (1024,1024)
                  const float* __restrict__ T,   // (16,1024,8)
                  float* __restrict__ Mg)        // (1024,128)
{
    const int lane   = threadIdx.x & 31;
    const int wave   = threadIdx.x >> 5;       // 0..7 -> 16-col tile
    const int mrow   = lane & 15;              // M (or N) index within tile
    const int khalf  = lane >> 4;              // selects K pair {0,1} vs {2,3}
    const int rowBase = blockIdx.x * 16;       // 64 row tiles
    const int colBase = wave * 16;

    // A source: row-major x, contiguous (K,K+1) -> one aligned float2 load
    const float* __restrict__ Arow = x + (size_t)(rowBase + mrow) * DDIM;

    // B source: column n = colBase + mrow maps to T[u= n>>3][d][r = n&7],
    // element (d, n) lives at T + u*D*R + d*R + r  (stride R=8 floats in d).
    const int n  = colBase + mrow;
    const float* __restrict__ Bcol = T + (size_t)(n >> 3) * (DDIM * ROW) + (n & 7);

    v8f c = {};  // f32 accumulator, 16x16 tile striped over the wave

    #pragma unroll 8
    for (int k0 = 0; k0 < DDIM; k0 += 4) {
        const int kk = k0 + 2 * khalf;
        // A fragment: global_load_b64 (8-byte aligned: kk is even)
        v2f a = *(const v2f*)(Arow + kk);
        // B fragment: two strided f32 gathers from T
        v2f b;
        b.x = Bcol[(size_t)kk * ROW];
        b.y = Bcol[(size_t)(kk + 1) * ROW];
        // D = A*B + C   (8 args: neg_a, A, neg_b, B, c_mod, C, reuse_a, reuse_b)
        c = __builtin_amdgcn_wmma_f32_16x16x4_f32(false, a, false, b,
                                                  (short)0, c, false, false);
    }

    // Store D per the C/D VGPR layout: VGPR v -> row M = v + 8*khalf, col = n
    float* __restrict__ Drow = Mg + (size_t)(rowBase + 8 * khalf) * NCOL + n;
    #pragma unroll
    for (int v = 0; v < 8; ++v)
        Drow[(size_t)v * NCOL] = c[v];
}

// ---------------------------------------------------------------------------
// Kernel 2: feats[i][u] = sum_j exp( -sum_r |M2[i][u][r] - M2[j][u][r]| )
// where the reference's raw reshape gives
//   M2[i][u][r] = Mg[(i%64)*16 + u][(i/64)*8 + r].
// One block per (u, 128-wide i-tile). The full slice M2[:,u,:] (1024x8 f32 =
// 32 KB) is staged once in LDS; inner j-loop reads are wave-uniform
// broadcasts. Writes feats straight into the output at column 1024+u.
// ---------------------------------------------------------------------------
__global__ void __launch_bounds__(128)
mbd_feats(const float* __restrict__ Mg,   // (1024,128)
          float* __restrict__ out)        // (1024,1040)
{
    __shared__ float lds[BATCH * ROW];    // 32 KB: M2[:, u, :]

    const int u = blockIdx.x;             // 0..15

    // Cooperative load of the shuffled slice into LDS
    for (int i = threadIdx.x; i < BATCH * ROW; i += 128) {
        const int bp  = i >> 3;           // i index (b')
        const int r   = i & 7;
        const int row = (bp & 63) * 16 + u;
        const int col = (bp >> 6) * 8 + r;
        lds[i] = Mg[(size_t)row * NCOL + col];
    }
    __syncthreads();

    const int bp = blockIdx.y * 128 + threadIdx.x;  // this thread's i

    float m[ROW];
    #pragma unroll
    for (int r = 0; r < ROW; ++r)
        m[r] = lds[bp * ROW + r];

    float acc = 0.0f;
    for (int j = 0; j < BATCH; ++j) {
        const float* __restrict__ mj = &lds[j * ROW];
        float s = 0.0f;
        #pragma unroll
        for (int r = 0; r < ROW; ++r)
            s += __builtin_fabsf(m[r] - mj[r]);
        acc += __expf(-s);                // v_exp_f32
    }

    out[(size_t)bp * OUTW + DDIM + u] = acc;
}

// ---------------------------------------------------------------------------
// Kernel 3: out[b][0:1024] = x[b][:]  (float4 copy; both bases 16B aligned)
// ---------------------------------------------------------------------------
__global__ void __launch_bounds__(256)
mbd_copy_x(const float* __restrict__ x, float* __restrict__ out)
{
    const int idx = blockIdx.x * blockDim.x + threadIdx.x;  // 0 .. 262143
    const int b   = idx >> 8;          // row
    const int c4  = idx & 255;         // float4 column
    const float4* src = (const float4*)(x   + (size_t)b * DDIM);
    float4*       dst = (float4*)      (out + (size_t)b * OUTW);
    dst[c4] = src[c4];
}

// ---------------------------------------------------------------------------
extern "C" void kernel_launch(void* const* d_in, const int* in_sizes, int n_in,
                              void* d_out, int out_size, void* d_ws, size_t ws_size,
                              hipStream_t stream)
{
    const float* x = (const float*)d_in[0];   // (1024,1024) f32
    const float* T = (const float*)d_in[1];   // (16,1024,8) f32
    float* out = (float*)d_out;               // (1024,1040) f32
    float* Mg  = (float*)d_ws;                // (1024,128) f32 = 512 KB scratch

    // 1) GEMM via f32 WMMA: 64 row-tile blocks x 8 waves (col tiles)
    mbd_gemm_wmma_f32<<<dim3(64), dim3(256), 0, stream>>>(x, T, Mg);

    // 2) Pairwise exp(-L1) features: (units, i-tiles) grid
    mbd_feats<<<dim3(UNITS, BATCH / 128), dim3(128), 0, stream>>>(Mg, out);

    // 3) Copy x into the first 1024 output columns
    mbd_copy_x<<<dim3((BATCH * DDIM / 4) / 256), dim3(256), 0, stream>>>(x, out);
}